// LMSLinear_57664230916914
// MI455X (gfx1250) — compile-verified
//
#include <hip/hip_runtime.h>
#include <stdint.h>

// Problem sizes (fixed by the reference)
#define B_    4
#define S_    4096
#define IN_   1024
#define OUT_  4096
#define RANK_ 32

// GEMM tiling
#define BM  128
#define BN  128
#define BK  32
#define LDP 36   // padded LDS row stride (floats): 144B rows -> 16B-aligned b128 LDS
                 // writes, and 16-lane fragment loads hit 16 distinct banks.
#define NPANELS (IN_ / BK)

typedef __attribute__((ext_vector_type(2))) float v2f;
typedef __attribute__((ext_vector_type(8))) float v8f;

// ---------------------------------------------------------------------------
// Kernel 0: w_eff[o][i] = weight[o][i] + sum_r w1[src][o][r] * w2[tgt][r][i]
// ---------------------------------------------------------------------------
__global__ __launch_bounds__(256)
void lms_weff_kernel(const float* __restrict__ weight,
                     const float* __restrict__ w1,
                     const float* __restrict__ w2,
                     const int*   __restrict__ src_id,
                     const int*   __restrict__ tgt_id,
                     float*       __restrict__ weff) {
    const int idx = blockIdx.x * 256 + threadIdx.x;   // < OUT_*IN_
    const int o = idx >> 10;          // / IN_
    const int i = idx & (IN_ - 1);    // % IN_
    const float* __restrict__ w1r = w1 + ((size_t)src_id[0] * OUT_ + o) * RANK_;
    const float* __restrict__ w2m = w2 + (size_t)tgt_id[0] * RANK_ * IN_ + i;
    float acc = 0.0f;
#pragma unroll
    for (int r = 0; r < RANK_; ++r)
        acc = fmaf(w1r[r], w2m[(size_t)r * IN_], acc);
    weff[idx] = weight[idx] + acc;
}

// ---------------------------------------------------------------------------
// CDNA5 async global->LDS copy (ASYNCcnt-tracked, no VGPR landing zone).
// GVS mode: 64-bit SGPR base + per-lane signed 32-bit byte offset; per-lane
// LDS destination byte address in a VGPR (low 32 bits of the generic pointer).
// ---------------------------------------------------------------------------
__device__ __forceinline__ void async_copy_b128(uint32_t lds_addr,
                                                const float* __restrict__ base,
                                                uint32_t voff_bytes) {
    asm volatile("global_load_async_to_lds_b128 %0, %1, %2"
                 :: "v"(lds_addr), "v"(voff_bytes), "s"(base)
                 : "memory");
}

__device__ __forceinline__ void wait_async_zero() {
    asm volatile("s_wait_asynccnt 0x0" ::: "memory");
}

// ---------------------------------------------------------------------------
// Kernel 1: out[m][n] = sum_k x[m][k] * weff[n][k] + bias[n]
// M=16384, N=4096, K=1024.  128x128 block tile, K-panel 32, double-buffered
// LDS filled by async-to-LDS DMA; 8 waves of 64x32; f32 WMMA 16x16x4.
// ---------------------------------------------------------------------------
__global__ __launch_bounds__(256)
void lms_gemm_kernel(const float* __restrict__ x,
                     const float* __restrict__ weff,
                     const float* __restrict__ bias,
                     float*       __restrict__ out) {
    __shared__ __align__(16) float As[2][BM][LDP];   // x rows   [m][k]
    __shared__ __align__(16) float Bs[2][BN][LDP];   // weff rows[n][k]

    const int tid  = threadIdx.x;
    const int lane = tid & 31;
    const int wave = tid >> 5;       // 0..7
    const int wm   = wave >> 2;      // 0..1  (M direction)
    const int wn   = wave & 3;       // 0..3  (N direction)
    const int lrow = lane & 15;      // 0..15
    const int lhi  = lane >> 4;      // 0..1

    const int m0 = blockIdx.y * BM;
    const int n0 = blockIdx.x * BN;

    // This thread's 4 staging slots (row, k-chunk) within a 128x32 panel.
    int s_row[4], s_c4[4];
#pragma unroll
    for (int i = 0; i < 4; ++i) {
        const int idx = i * 256 + tid;        // 0..1023
        s_row[i] = idx >> 3;                  // 0..127
        s_c4[i]  = (idx & 7) << 2;            // 0,4,...,28 (floats)
    }

    v8f acc[4][2] = {};              // 64x32 per wave = 8 fragments

    // ---- issue panel 0 into buffer 0 --------------------------------------
#pragma unroll
    for (int i = 0; i < 4; ++i) {
        async_copy_b128((uint32_t)(uintptr_t)&As[0][s_row[i]][s_c4[i]], x,
                        (uint32_t)(((m0 + s_row[i]) * IN_ + s_c4[i]) * 4));
        async_copy_b128((uint32_t)(uintptr_t)&Bs[0][s_row[i]][s_c4[i]], weff,
                        (uint32_t)(((n0 + s_row[i]) * IN_ + s_c4[i]) * 4));
    }
    wait_async_zero();
    __syncthreads();

    for (int p = 0; p < NPANELS; ++p) {
        const int buf = p & 1;
        const int kp  = p * BK;

        // ---- prefetch next panel into the other buffer (async DMA) --------
        if (p + 1 < NPANELS) {
            const int kn = kp + BK;
#pragma unroll
            for (int i = 0; i < 4; ++i) {
                async_copy_b128(
                    (uint32_t)(uintptr_t)&As[buf ^ 1][s_row[i]][s_c4[i]], x,
                    (uint32_t)(((m0 + s_row[i]) * IN_ + kn + s_c4[i]) * 4));
                async_copy_b128(
                    (uint32_t)(uintptr_t)&Bs[buf ^ 1][s_row[i]][s_c4[i]], weff,
                    (uint32_t)(((n0 + s_row[i]) * IN_ + kn + s_c4[i]) * 4));
            }
        }

        // ---- compute on current buffer ------------------------------------
#pragma unroll
        for (int kk = 0; kk < BK; kk += 4) {
            // A fragment (16x4 f32): lane l holds A[l&15][(l>>4)*2 + j], j=0,1
            v2f a[4];
#pragma unroll
            for (int mt = 0; mt < 4; ++mt)
                a[mt] = *(const v2f*)&As[buf][wm * 64 + mt * 16 + lrow][kk + lhi * 2];
            // B fragment (4x16 f32): lane l holds B[(l>>4)*2 + j][l&15]
            v2f b[2];
#pragma unroll
            for (int nt = 0; nt < 2; ++nt)
                b[nt] = *(const v2f*)&Bs[buf][wn * 32 + nt * 16 + lrow][kk + lhi * 2];

#pragma unroll
            for (int mt = 0; mt < 4; ++mt)
#pragma unroll
                for (int nt = 0; nt < 2; ++nt)
                    acc[mt][nt] = __builtin_amdgcn_wmma_f32_16x16x4_f32(
                        /*neg_a=*/false, a[mt],
                        /*neg_b=*/false, b[nt],
                        /*c_mod=*/(short)0, acc[mt][nt],
                        /*reuse_a=*/false, /*reuse_b=*/false);
        }

        // Next panel must have landed (this wave) and everyone must be done
        // reading the current buffer before it is overwritten next iteration.
        wait_async_zero();
        __syncthreads();
    }

    // Epilogue: add bias, write out.  C/D layout: vgpr r, lane l ->
    // row = r + (l>>4)*8, col = l&15 within each 16x16 tile.
    float bv[2];
#pragma unroll
    for (int nt = 0; nt < 2; ++nt)
        bv[nt] = bias[n0 + wn * 32 + nt * 16 + lrow];

#pragma unroll
    for (int mt = 0; mt < 4; ++mt) {
#pragma unroll
        for (int nt = 0; nt < 2; ++nt) {
            const int col = n0 + wn * 32 + nt * 16 + lrow;
#pragma unroll
            for (int r = 0; r < 8; ++r) {
                const int row = m0 + wm * 64 + mt * 16 + lhi * 8 + r;
                out[(size_t)row * OUT_ + col] = acc[mt][nt][r] + bv[nt];
            }
        }
    }
}

// ---------------------------------------------------------------------------
// Host launch.  Inputs (setup_inputs order):
//   0: x [B,S,IN] f32   1: weight [OUT,IN] f32   2: bias [OUT] f32
//   3: w1 [NLANG,OUT,RANK] f32   4: w2 [NLANG,RANK,IN] f32
//   5: src_id (int)     6: tgt_id (int)
// d_ws: needs OUT_*IN_*4 = 16 MB for w_eff.
// ---------------------------------------------------------------------------
extern "C" void kernel_launch(void* const* d_in, const int* in_sizes, int n_in,
                              void* d_out, int out_size, void* d_ws, size_t ws_size,
                              hipStream_t stream) {
    const float* x      = (const float*)d_in[0];
    const float* weight = (const float*)d_in[1];
    const float* bias   = (const float*)d_in[2];
    const float* w1     = (const float*)d_in[3];
    const float* w2     = (const float*)d_in[4];
    const int*   src    = (const int*)d_in[5];
    const int*   tgt    = (const int*)d_in[6];
    float* out  = (float*)d_out;
    float* weff = (float*)d_ws;     // 16 MB scratch

    lms_weff_kernel<<<(OUT_ * IN_) / 256, 256, 0, stream>>>(
        weight, w1, w2, src, tgt, weff);

    dim3 grid(OUT_ / BN, (B_ * S_) / BM);   // n-blocks fastest: weff stays hot in L2
    lms_gemm_kernel<<<grid, 256, 0, stream>>>(x, weff, bias, out);
}